// MDMF_66228395704409
// MI455X (gfx1250) — compile-verified
//
#include <hip/hip_runtime.h>
#include <hip/hip_bf16.h>
#include <math.h>

typedef __attribute__((ext_vector_type(2))) float v2f;
typedef __attribute__((ext_vector_type(8))) float v8f;

#define MDIM 8192
#define NDIM 4096
#define KDIM 128
#define LAMBDA_REG 0.01

// ---------------------------------------------------------------------------
// acc slots (doubles in d_ws):
//  0: sum((U V^T - A)^2)
//  1: sum((U U^T - S_m)^2)
//  2: sum((V V^T - S_d)^2)
//  3: sum(U^2)
//  4: sum(V^2)
// ---------------------------------------------------------------------------

__global__ void mdmf_init(double* acc) {
    int t = threadIdx.x;
    if (t < 8) acc[t] = 0.0;
}

// Generic fused kernel: accumulate sum over the (R x C) matrix of
// (X Y^T - T)^2, where X is (R x 128), Y is (C x 128), T is (R x C).
// Each wave computes one 32x32 output tile via V_WMMA_F32_16X16X4_F32.
__global__ __launch_bounds__(256) void mdmf_gemm_sq(
    const float* __restrict__ X,
    const float* __restrict__ Y,
    const float* __restrict__ T,
    double* __restrict__ acc,
    int Ccols,            // number of columns of T (and rows of Y)
    int tilesPerRow)      // Ccols / 32
{
    const int tid  = threadIdx.x;
    const int wave = tid >> 5;
    const int lane = tid & 31;
    const int laneM = lane & 15;   // row/col within 16-tile
    const int hi    = lane >> 4;   // upper half-wave flag

    const int g      = blockIdx.x * 8 + wave;   // global 32x32 tile id
    const int tile_r = g / tilesPerRow;
    const int tile_c = g % tilesPerRow;
    const int row0   = tile_r * 32;
    const int col0   = tile_c * 32;

    // A-matrix (16x4 f32) per-lane layout:
    //   lanes 0-15  : {K=4k+0, K=4k+1} for M = lane
    //   lanes 16-31 : {K=4k+2, K=4k+3} for M = lane-16
    // B-matrix (4x16) of Y^T has the identical per-lane formula over Y rows.
    const int koff = hi * 2;
    const float* x0 = X + (size_t)(row0 + laneM)      * KDIM + koff;
    const float* x1 = X + (size_t)(row0 + 16 + laneM) * KDIM + koff;
    const float* y0 = Y + (size_t)(col0 + laneM)      * KDIM + koff;
    const float* y1 = Y + (size_t)(col0 + 16 + laneM) * KDIM + koff;

    v8f c00 = {}; v8f c01 = {}; v8f c10 = {}; v8f c11 = {};

#pragma unroll 8
    for (int k = 0; k < KDIM; k += 4) {
        v2f a0 = *(const v2f*)(x0 + k);
        v2f a1 = *(const v2f*)(x1 + k);
        v2f b0 = *(const v2f*)(y0 + k);
        v2f b1 = *(const v2f*)(y1 + k);
        c00 = __builtin_amdgcn_wmma_f32_16x16x4_f32(false, a0, false, b0,
                                                    (short)0, c00, false, false);
        c01 = __builtin_amdgcn_wmma_f32_16x16x4_f32(false, a0, false, b1,
                                                    (short)0, c01, false, false);
        c10 = __builtin_amdgcn_wmma_f32_16x16x4_f32(false, a1, false, b0,
                                                    (short)0, c10, false, false);
        c11 = __builtin_amdgcn_wmma_f32_16x16x4_f32(false, a1, false, b1,
                                                    (short)0, c11, false, false);
    }

    // C/D 16x16 f32 layout: VGPR i, lanes 0-15 -> (M=i,     N=lane),
    //                               lanes 16-31 -> (M=i+8, N=lane-16).
    // Stream T exactly once with non-temporal hint (keep U/V hot in L2).
    float local = 0.0f;
    {
        const int colA = col0 + laneM;        // tj = 0
        const int colB = col0 + 16 + laneM;   // tj = 1
        const int rTop = row0 + hi * 8;       // ti = 0
        const int rBot = row0 + 16 + hi * 8;  // ti = 1
#pragma unroll
        for (int i = 0; i < 8; ++i) {
            const size_t offT = (size_t)(rTop + i) * Ccols;
            const size_t offB = (size_t)(rBot + i) * Ccols;
            float t00 = __builtin_nontemporal_load(&T[offT + colA]);
            float t01 = __builtin_nontemporal_load(&T[offT + colB]);
            float t10 = __builtin_nontemporal_load(&T[offB + colA]);
            float t11 = __builtin_nontemporal_load(&T[offB + colB]);
            float d0 = c00[i] - t00;
            float d1 = c01[i] - t01;
            float d2 = c10[i] - t10;
            float d3 = c11[i] - t11;
            local += d0 * d0 + d1 * d1 + d2 * d2 + d3 * d3;
        }
    }

    // Block reduction -> one f64 atomic per block.
    __shared__ float sred[256];
    sred[tid] = local;
    __syncthreads();
#pragma unroll
    for (int s = 128; s > 0; s >>= 1) {
        if (tid < s) sred[tid] += sred[tid + s];
        __syncthreads();
    }
    if (tid == 0) atomicAdd(acc, (double)sred[0]);
}

// Simple grid-stride sum of squares (for ||U||_F^2, ||V||_F^2).
__global__ __launch_bounds__(256) void mdmf_sumsq(
    const float* __restrict__ x, int n, double* __restrict__ acc)
{
    float local = 0.0f;
    for (int i = blockIdx.x * 256 + threadIdx.x; i < n; i += gridDim.x * 256) {
        float v = x[i];
        local += v * v;
    }
    __shared__ float sred[256];
    sred[threadIdx.x] = local;
    __syncthreads();
#pragma unroll
    for (int s = 128; s > 0; s >>= 1) {
        if (threadIdx.x < s) sred[threadIdx.x] += sred[threadIdx.x + s];
        __syncthreads();
    }
    if (threadIdx.x == 0) atomicAdd(acc, (double)sred[0]);
}

__global__ void mdmf_finalize(const double* __restrict__ acc,
                              float* __restrict__ out)
{
    if (threadIdx.x == 0 && blockIdx.x == 0) {
        double recon = acc[0] / ((double)MDIM * (double)NDIM);
        double reg   = sqrt(acc[3]) + sqrt(acc[4]) + sqrt(acc[1]) + sqrt(acc[2]);
        out[0] = (float)(recon + LAMBDA_REG * reg);
    }
}

extern "C" void kernel_launch(void* const* d_in, const int* in_sizes, int n_in,
                              void* d_out, int out_size, void* d_ws, size_t ws_size,
                              hipStream_t stream) {
    const float* A  = (const float*)d_in[0];   // [M, N]
    const float* Sm = (const float*)d_in[1];   // [M, M]
    const float* Sd = (const float*)d_in[2];   // [N, N]
    const float* U  = (const float*)d_in[3];   // [M, K]
    const float* V  = (const float*)d_in[4];   // [N, K]
    float* out  = (float*)d_out;
    double* acc = (double*)d_ws;

    mdmf_init<<<1, 32, 0, stream>>>(acc);

    // recon: (U V^T - A)^2  -> acc[0]
    {
        int tiles = (MDIM / 32) * (NDIM / 32);         // 32768
        mdmf_gemm_sq<<<tiles / 8, 256, 0, stream>>>(U, V, A, acc + 0,
                                                    NDIM, NDIM / 32);
    }
    // (U U^T - S_m)^2 -> acc[1]
    {
        int tiles = (MDIM / 32) * (MDIM / 32);         // 65536
        mdmf_gemm_sq<<<tiles / 8, 256, 0, stream>>>(U, U, Sm, acc + 1,
                                                    MDIM, MDIM / 32);
    }
    // (V V^T - S_d)^2 -> acc[2]
    {
        int tiles = (NDIM / 32) * (NDIM / 32);         // 16384
        mdmf_gemm_sq<<<tiles / 8, 256, 0, stream>>>(V, V, Sd, acc + 2,
                                                    NDIM, NDIM / 32);
    }
    // ||U||^2, ||V||^2
    mdmf_sumsq<<<256, 256, 0, stream>>>(U, MDIM * KDIM, acc + 3);
    mdmf_sumsq<<<256, 256, 0, stream>>>(V, NDIM * KDIM, acc + 4);

    mdmf_finalize<<<1, 32, 0, stream>>>(acc, out);
}